// BertSelfAttention_59244778881093
// MI455X (gfx1250) — compile-verified
//
#include <hip/hip_runtime.h>
#include <hip/hip_bf16.h>
#include <math.h>

typedef __attribute__((ext_vector_type(2))) float v2f;
typedef __attribute__((ext_vector_type(8))) float v8f;

#define HDIM 256
#define NHEADS 8
#define DHEAD 32
#define DNBR 16

// ---------------------------------------------------------------------------
// Phase 1: Q/K/V projections, out[n,j] = sum_i h[n,i]*W[j,i] + b[j]
// One wave computes one 16x16 tile of one projection using
// V_WMMA_F32_16X16X4_F32, accumulating over H=256 in steps of 4.
//
// f32 WMMA layouts (ISA 7.12.2):
//   A 16x4:  lanes 0-15 -> M=lane, {v0,v1}={K0,K1}; lanes 16-31 -> {K2,K3}
//   B 4x16:  lanes 0-15 -> N=lane, {v0,v1}={K0,K1}; lanes 16-31 -> {K2,K3}
//   C/D 16x16: VGPR r, lanes 0-15 -> M=r,N=lane; lanes 16-31 -> M=r+8
// ---------------------------------------------------------------------------
__global__ __launch_bounds__(128) void qkv_proj_wmma(
    const float* __restrict__ h,
    const float* __restrict__ Wq, const float* __restrict__ bq,
    const float* __restrict__ Wk, const float* __restrict__ bk,
    const float* __restrict__ Wv, const float* __restrict__ bv,
    float* __restrict__ Q, float* __restrict__ K, float* __restrict__ V,
    int N)
{
    const int lane  = threadIdx.x & 31;
    const int wave  = threadIdx.x >> 5;
    const int tileM = blockIdx.x;               // rows of h (16 per tile)
    const int tileN = blockIdx.y * 4 + wave;    // output cols (16 per tile)

    const float* W; const float* bias; float* out;
    if (blockIdx.z == 0)      { W = Wq; bias = bq; out = Q; }
    else if (blockIdx.z == 1) { W = Wk; bias = bk; out = K; }
    else                      { W = Wv; bias = bv; out = V; }

    const int mn    = lane & 15;       // M index for A, N index for B
    const int khalf = (lane >> 4) * 2; // 0 or 2: which K-pair this lane holds

    const int rowA   = min(tileM * 16 + mn, N - 1);   // clamp keeps EXEC full
    const int colB   = tileN * 16 + mn;

    const float* aPtr = h + (size_t)rowA * HDIM + khalf;
    const float* bPtr = W + (size_t)colB * HDIM + khalf;  // B[k][n] = W[n][k]

    v8f c = {};
#pragma unroll 8
    for (int kk = 0; kk < HDIM; kk += 4) {
        v2f a = *(const v2f*)(aPtr + kk);
        v2f b = *(const v2f*)(bPtr + kk);
        c = __builtin_amdgcn_wmma_f32_16x16x4_f32(
                /*neg_a=*/false, a, /*neg_b=*/false, b,
                /*c_mod=*/(short)0, c, /*reuse_a=*/false, /*reuse_b=*/false);
    }

    // Epilogue: add bias, store D. Lanes 0-15 cover M=r, lanes 16-31 M=r+8.
    const int outCol = tileN * 16 + mn;
    const float bval = bias[outCol];
    const int mBase  = tileM * 16 + ((lane >> 4) << 3);
    float* orow = out + (size_t)mBase * HDIM + outCol;

    if (tileM * 16 + 16 <= N) {
        // Full tile (the only case when N % 16 == 0): unguarded coalesced stores.
#pragma unroll
        for (int r = 0; r < 8; ++r)
            orow[(size_t)r * HDIM] = c[r] + bval;
    } else {
#pragma unroll
        for (int r = 0; r < 8; ++r)
            if (mBase + r < N) orow[(size_t)r * HDIM] = c[r] + bval;
    }
}

// ---------------------------------------------------------------------------
// Phase 2: per-(node, head) attention over 16 gathered neighbors.
// Q/K/V rows are hot in L2 (20.5 MB each). dh=32 -> 8 float4 per row-segment.
// ---------------------------------------------------------------------------
__global__ __launch_bounds__(256) void neighbor_attn(
    const float* __restrict__ Q, const float* __restrict__ K,
    const float* __restrict__ V, const float* __restrict__ mask,
    const int* __restrict__ nbr, float* __restrict__ out, int N)
{
    const int t = blockIdx.x * blockDim.x + threadIdx.x;
    const int n    = t >> 3;   // node
    const int head = t & 7;    // head
    if (n >= N) return;

    const float inv_sqrt_dh = 0.1767766952966369f; // 1/sqrt(32)

    // Load this head's query (32 floats).
    float q[DHEAD];
    {
        const float4* qp = (const float4*)(Q + (size_t)n * HDIM + head * DHEAD);
#pragma unroll
        for (int j = 0; j < DHEAD / 4; ++j) {
            float4 v4 = qp[j];
            q[4*j+0] = v4.x; q[4*j+1] = v4.y; q[4*j+2] = v4.z; q[4*j+3] = v4.w;
        }
    }

    int idx[DNBR];
#pragma unroll
    for (int d = 0; d < DNBR; ++d) idx[d] = nbr[(size_t)n * DNBR + d];

    // Scores
    float s[DNBR];
#pragma unroll
    for (int d = 0; d < DNBR; ++d) {
        const float4* kp = (const float4*)(K + (size_t)idx[d] * HDIM + head * DHEAD);
        float acc = 0.f;
#pragma unroll
        for (int j = 0; j < DHEAD / 4; ++j) {
            float4 kv = kp[j];
            acc += q[4*j+0] * kv.x + q[4*j+1] * kv.y
                 + q[4*j+2] * kv.z + q[4*j+3] * kv.w;
        }
        s[d] = acc * inv_sqrt_dh + mask[idx[d]];
    }

    // Softmax over the 16 neighbors.
    float m = s[0];
#pragma unroll
    for (int d = 1; d < DNBR; ++d) m = fmaxf(m, s[d]);
    float sum = 0.f;
#pragma unroll
    for (int d = 0; d < DNBR; ++d) { s[d] = __expf(s[d] - m); sum += s[d]; }
    const float rsum = 1.f / sum;

    // Weighted sum of V.
    float ctx[DHEAD];
#pragma unroll
    for (int j = 0; j < DHEAD; ++j) ctx[j] = 0.f;
#pragma unroll
    for (int d = 0; d < DNBR; ++d) {
        const float p = s[d] * rsum;
        const float4* vp = (const float4*)(V + (size_t)idx[d] * HDIM + head * DHEAD);
#pragma unroll
        for (int j = 0; j < DHEAD / 4; ++j) {
            float4 vv = vp[j];
            ctx[4*j+0] += p * vv.x; ctx[4*j+1] += p * vv.y;
            ctx[4*j+2] += p * vv.z; ctx[4*j+3] += p * vv.w;
        }
    }

    float4* op = (float4*)(out + (size_t)n * HDIM + head * DHEAD);
#pragma unroll
    for (int j = 0; j < DHEAD / 4; ++j) {
        float4 v4; v4.x = ctx[4*j+0]; v4.y = ctx[4*j+1];
        v4.z = ctx[4*j+2]; v4.w = ctx[4*j+3];
        op[j] = v4;
    }
}

extern "C" void kernel_launch(void* const* d_in, const int* in_sizes, int n_in,
                              void* d_out, int out_size, void* d_ws, size_t ws_size,
                              hipStream_t stream) {
    const float* h    = (const float*)d_in[0];
    const float* mask = (const float*)d_in[1];
    const int*   nbr  = (const int*)d_in[2];
    const float* Wq   = (const float*)d_in[3];
    const float* bq   = (const float*)d_in[4];
    const float* Wk   = (const float*)d_in[5];
    const float* bk   = (const float*)d_in[6];
    const float* Wv   = (const float*)d_in[7];
    const float* bv   = (const float*)d_in[8];
    float*       out  = (float*)d_out;

    const int N = in_sizes[0] / HDIM;

    float* Q = (float*)d_ws;
    float* K = Q + (size_t)N * HDIM;
    float* V = K + (size_t)N * HDIM;

    // Phase 1: three 256-wide projections via f32 WMMA.
    // grid: (row tiles, 4 col-tile groups, 3 matrices); 4 waves/block.
    dim3 g1((N + 15) / 16, (HDIM / 16) / 4, 3);
    qkv_proj_wmma<<<g1, 128, 0, stream>>>(h, Wq, bq, Wk, bk, Wv, bv, Q, K, V, N);

    // Phase 2: one thread per (node, head).
    const int threads = N * NHEADS;
    neighbor_attn<<<(threads + 255) / 256, 256, 0, stream>>>(Q, K, V, mask, nbr, out, N);
}